// EmbeddingAlignmentGNN_81793357185798
// MI455X (gfx1250) — compile-verified
//
#include <hip/hip_runtime.h>

// ---------------------------------------------------------------------------
// GraphSAGE 3-layer GNN for MI455X (gfx1250, wave32, WMMA)
//   N=100000 nodes, E=800000 edges, D=128
// - Dense GEMMs: v_wmma_f32_16x16x4_f32 (fp32 matrix path).
// - A-tiles staged via GLOBAL_LOAD_ASYNC_TO_LDS_B128 (ASYNCcnt path);
//   deg_inv mean-scaling folded into the epilogue (dual accumulators).
// - Scatter-mean: wave-per-edge float4 gathers + f32 atomics (L2-resident:
//   51MB feature matrix << 192MB L2).
// ---------------------------------------------------------------------------

#define GN 100000
#define GE 800000
#define GD 128

typedef __attribute__((ext_vector_type(2))) float v2f;
typedef __attribute__((ext_vector_type(4))) float v4f;
typedef __attribute__((ext_vector_type(8))) float v8f;

// ---------------------------------------------------------- async helpers
__device__ __forceinline__ void async_wait0() {
#if __has_builtin(__builtin_amdgcn_s_wait_asynccnt)
    __builtin_amdgcn_s_wait_asynccnt(0);
#else
    asm volatile("s_wait_asynccnt 0x0" ::: "memory");
#endif
}

// Copy 16 bytes per lane: global -> LDS, tracked by ASYNCcnt.
__device__ __forceinline__ void async_copy_b128(void* lds_dst, const void* gsrc) {
    // Low 32 bits of a generic shared pointer == wave-relative LDS byte addr.
    unsigned lds_off = (unsigned)(unsigned long long)lds_dst;
    asm volatile("global_load_async_to_lds_b128 %0, %1, off"
                 :: "v"(lds_off), "v"(gsrc)
                 : "memory");
}

// ---------------------------------------------------------------- zero fill
__global__ __launch_bounds__(256) void zero_k(v4f* __restrict__ p, long n4) {
    long i = (long)blockIdx.x * blockDim.x + threadIdx.x;
    long stride = (long)gridDim.x * blockDim.x;
    v4f z = {0.f, 0.f, 0.f, 0.f};
    for (; i < n4; i += stride) p[i] = z;
}

// ---------------------------------------------------------------- degree
__global__ __launch_bounds__(256) void degree_k(const long long* __restrict__ dst,
                                                float* __restrict__ deg, int ne) {
    int e = blockIdx.x * blockDim.x + threadIdx.x;
    if (e < ne) atomicAdd(&deg[(int)dst[e]], 1.0f);
}

__global__ __launch_bounds__(256) void deginv_k(float* __restrict__ deg, int n) {
    int i = blockIdx.x * blockDim.x + threadIdx.x;
    if (i < n) {
        float d = deg[i];
        deg[i] = (d > 0.f) ? (1.0f / d) : 0.0f;
    }
}

// ------------------------------------------------- scatter-add (wave/edge)
// One wave32 per edge: 32 lanes x float4 = 128 floats (one feature row).
__global__ __launch_bounds__(256) void scatter_add_k(const float* __restrict__ h,
                                                     const long long* __restrict__ src,
                                                     const long long* __restrict__ dst,
                                                     float* __restrict__ agg, int ne) {
    int t = blockIdx.x * blockDim.x + threadIdx.x;
    int e = t >> 5;
    if (e >= ne) return;
    int lane = t & 31;
    long long s = src[e];
    long long d = dst[e];
    const v4f* row = (const v4f*)(h + (size_t)s * GD);
    v4f v = row[lane];
    float* o = agg + (size_t)d * GD + lane * 4;
    atomicAdd(o + 0, v.x);
    atomicAdd(o + 1, v.y);
    atomicAdd(o + 2, v.z);
    atomicAdd(o + 3, v.w);
}

// ------------------------------------------------------------- fused GEMM
// Per block: 16 rows x 128 cols of
//   maybe_relu( deg_inv*(agg @ B1^T) + bias + A2 @ B2^T ) + maybe identity
// 8 waves/block, one 16x16 tile per wave, K-loop of v_wmma_f32_16x16x4_f32.
// A tiles (16x128 fp32 each) staged via async global->LDS copies.
__global__ __launch_bounds__(256) void sage_gemm_k(const float* __restrict__ A1,      // agg or null
                                                   const float* __restrict__ deginv,  // [N] or null
                                                   const float* __restrict__ B1,      // W_l [D,D] or null
                                                   const float* __restrict__ A2,      // h   [N,D]
                                                   const float* __restrict__ B2,      // W_r [D,D]
                                                   const float* __restrict__ bias,    // [D] or null
                                                   const float* __restrict__ identity,// [N,D] or null
                                                   float* __restrict__ out,           // [N,D]
                                                   int do_relu) {
    __shared__ float sA1[16 * GD];
    __shared__ float sA2[16 * GD];

    const int tid = threadIdx.x;
    const int m0  = blockIdx.x * 16;

    // Stage A tiles: 512 x b128 per tile, 2 per thread, direct to LDS.
    {
        const v4f* g2 = (const v4f*)(A2 + (size_t)m0 * GD);
        v4f* s2 = (v4f*)sA2;
        async_copy_b128(&s2[tid],       &g2[tid]);
        async_copy_b128(&s2[tid + 256], &g2[tid + 256]);
        if (A1 != nullptr) {
            const v4f* g1 = (const v4f*)(A1 + (size_t)m0 * GD);
            v4f* s1 = (v4f*)sA1;
            async_copy_b128(&s1[tid],       &g1[tid]);
            async_copy_b128(&s1[tid + 256], &g1[tid + 256]);
        }
        async_wait0();
    }
    __syncthreads();

    const int wave = tid >> 5;       // 0..7 -> N tile
    const int lane = tid & 31;
    const int half = lane >> 4;      // 0: K+0/1, 1: K+2/3 (f32 WMMA layout)
    const int lm   = lane & 15;      // A row / B col within tile
    const int n0   = wave * 16;

    // Weight rows are hot in cache; hint prefetch (global_prefetch_b8).
    __builtin_prefetch(B2 + (size_t)(n0 + lm) * GD, 0, 3);
    if (B1 != nullptr) __builtin_prefetch(B1 + (size_t)(n0 + lm) * GD, 0, 3);

    v8f c1 = {};   // agg @ W_l^T (unscaled; deg_inv applied per-row in epilogue)
    v8f c2 = {};   // h   @ W_r^T

    if (A1 != nullptr) {
        const float* arow = sA1 + lm * GD;
        const float* wrow = B1 + (size_t)(n0 + lm) * GD;
#pragma unroll 8
        for (int k = 0; k < GD; k += 4) {
            v2f a = *(const v2f*)(arow + k + half * 2);   // A: M=lm, K=k+2*half(+1)
            v2f b = *(const v2f*)(wrow + k + half * 2);   // B[k][n] = W[n][k]
            c1 = __builtin_amdgcn_wmma_f32_16x16x4_f32(false, a, false, b,
                                                       (short)0, c1, false, false);
        }
    }
    {
        const float* arow = sA2 + lm * GD;
        const float* wrow = B2 + (size_t)(n0 + lm) * GD;
#pragma unroll 8
        for (int k = 0; k < GD; k += 4) {
            v2f a = *(const v2f*)(arow + k + half * 2);
            v2f b = *(const v2f*)(wrow + k + half * 2);
            c2 = __builtin_amdgcn_wmma_f32_16x16x4_f32(false, a, false, b,
                                                       (short)0, c2, false, false);
        }
    }

    // Epilogue: C/D layout -> lane col n = n0+lm, VGPR j row m = j + 8*half.
    const int n = n0 + lm;
    const float bv = (bias != nullptr) ? bias[n] : 0.0f;
#pragma unroll
    for (int j = 0; j < 8; ++j) {
        const int m = m0 + j + 8 * half;
        float v = c2[j] + bv;
        if (A1 != nullptr) v += deginv[m] * c1[j];
        if (do_relu) v = fmaxf(v, 0.0f);
        if (identity != nullptr) v += identity[(size_t)m * GD + n];
        out[(size_t)m * GD + n] = v;
    }
}

// ---------------------------------------------------------- L2 normalize
__global__ __launch_bounds__(128) void normalize_k(const float* __restrict__ h,
                                                   float* __restrict__ out) {
    __shared__ float s[128];
    const int row = blockIdx.x;
    const int tid = threadIdx.x;
    float v = h[(size_t)row * GD + tid];
    s[tid] = v * v;
    __syncthreads();
#pragma unroll
    for (int off = 64; off > 0; off >>= 1) {
        if (tid < off) s[tid] += s[tid + off];
        __syncthreads();
    }
    float norm = sqrtf(s[0]);
    out[(size_t)row * GD + tid] = v / fmaxf(norm, 1e-12f);
}

// ---------------------------------------------------------------------------
extern "C" void kernel_launch(void* const* d_in, const int* in_sizes, int n_in,
                              void* d_out, int out_size, void* d_ws, size_t ws_size,
                              hipStream_t stream) {
    const float*     x   = (const float*)d_in[0];
    const long long* ei  = (const long long*)d_in[1];   // int64 [2,E]
    const float*     Wp  = (const float*)d_in[2];
    const float*     W1l = (const float*)d_in[3];
    const float*     b1  = (const float*)d_in[4];
    const float*     W1r = (const float*)d_in[5];
    const float*     W2l = (const float*)d_in[6];
    const float*     b2  = (const float*)d_in[7];
    const float*     W2r = (const float*)d_in[8];
    const float*     W3l = (const float*)d_in[9];
    const float*     b3  = (const float*)d_in[10];
    const float*     W3r = (const float*)d_in[11];
    float*           out = (float*)d_out;

    const long long* src = ei;        // edge_index[0]
    const long long* dst = ei + GE;   // edge_index[1]

    float* deg  = (float*)d_ws;                 // N floats (becomes deg_inv)
    float* bufA = deg + GN;                     // N*D
    float* bufB = bufA + (size_t)GN * GD;       // N*D
    float* agg  = bufB + (size_t)GN * GD;       // N*D

    const int mtiles = GN / 16;                 // 6250, exact
    const long nd4   = (long)GN * GD / 4;
    const int  sgrid = (GE * 32) / 256;         // 100000 blocks, exact

    // degrees
    zero_k<<<128, 256, 0, stream>>>((v4f*)deg, (long)GN / 4);
    degree_k<<<(GE + 255) / 256, 256, 0, stream>>>(dst, deg, GE);
    deginv_k<<<(GN + 255) / 256, 256, 0, stream>>>(deg, GN);

    // projection: bufA = x @ Wp^T
    sage_gemm_k<<<mtiles, 256, 0, stream>>>(nullptr, nullptr, nullptr,
                                            x, Wp, nullptr, nullptr, bufA, 0);

    // layer 1: bufB = relu(mean(bufA)@W1l^T + b1 + bufA@W1r^T) + x
    zero_k<<<4096, 256, 0, stream>>>((v4f*)agg, nd4);
    scatter_add_k<<<sgrid, 256, 0, stream>>>(bufA, src, dst, agg, GE);
    sage_gemm_k<<<mtiles, 256, 0, stream>>>(agg, deg, W1l, bufA, W1r, b1, x, bufB, 1);

    // layer 2: bufA = relu(mean(bufB)@W2l^T + b2 + bufB@W2r^T) + bufB
    zero_k<<<4096, 256, 0, stream>>>((v4f*)agg, nd4);
    scatter_add_k<<<sgrid, 256, 0, stream>>>(bufB, src, dst, agg, GE);
    sage_gemm_k<<<mtiles, 256, 0, stream>>>(agg, deg, W2l, bufB, W2r, b2, bufB, bufA, 1);

    // layer 3: bufB = mean(bufA)@W3l^T + b3 + bufA@W3r^T
    zero_k<<<4096, 256, 0, stream>>>((v4f*)agg, nd4);
    scatter_add_k<<<sgrid, 256, 0, stream>>>(bufA, src, dst, agg, GE);
    sage_gemm_k<<<mtiles, 256, 0, stream>>>(agg, deg, W3l, bufA, W3r, b3, nullptr, bufB, 0);

    // row-wise L2 normalize -> out
    normalize_k<<<GN, 128, 0, stream>>>(bufB, out);
}